// SparsityLoss_9500467658939
// MI455X (gfx1250) — compile-verified
//
#include <hip/hip_runtime.h>

// CDNA5 / gfx1250: wave32, WMMA 16x16 shapes.
typedef __attribute__((ext_vector_type(2))) float v2f;
typedef __attribute__((ext_vector_type(8))) float v8f;

// Wave-level sum using V_WMMA_F32_16X16X4_F32:
//   A = ones (16x4), B = per-lane partials (4x16, 2 VGPRs/lane = 64 values),
//   D[m][n] = sum_k B[k][n]  -> 16 column sums, finished with 4 shfl_xor steps.
// EXEC must be all ones at the call site (callers guarantee reconvergence).
__device__ __forceinline__ float wave_sum_wmma(float s) {
  v2f a; a[0] = 1.0f; a[1] = 1.0f;   // all-ones A operand
  v2f b; b[0] = s;    b[1] = 0.0f;   // lane partial in first slot
  v8f c = {};
  // 8 args: (neg_a, A, neg_b, B, c_mod, C, reuse_a, reuse_b)
  c = __builtin_amdgcn_wmma_f32_16x16x4_f32(
      /*neg_a=*/false, a, /*neg_b=*/false, b,
      /*c_mod=*/(short)0, c, /*reuse_a=*/false, /*reuse_b=*/false);
  float r = c[0];  // lane L holds column sum for N = L % 16 (rows duplicated)
  // Sum the 16 column sums; xor masks 1..8 stay within each 16-lane half,
  // so every lane ends up holding the full wave total.
  #pragma unroll
  for (int off = 1; off < 16; off <<= 1)
    r += __shfl_xor(r, off, 32);
  return r;
}

// Pass 1: bandwidth-bound masked sum. b128 loads of masks, b32 loads of the
// 4-byte valid groups, grid-stride, one f32 partial per block (deterministic,
// no atomics -> graph-replay safe without workspace re-init).
__global__ void SparsityLoss_partial_kernel(const float4* __restrict__ m,
                                            const uchar4* __restrict__ v,
                                            float* __restrict__ partials,
                                            int n4) {
  __shared__ float smem[8];  // 256 threads = 8 waves on wave32
  const int tid = threadIdx.x;
  const int stride = gridDim.x * blockDim.x;
  float4 acc = make_float4(0.f, 0.f, 0.f, 0.f);
  for (int i = blockIdx.x * blockDim.x + tid; i < n4; i += stride) {
    float4 x = m[i];
    uchar4 b = v[i];
    acc.x += b.x ? x.x : 0.0f;
    acc.y += b.y ? x.y : 0.0f;
    acc.z += b.z ? x.z : 0.0f;
    acc.w += b.w ? x.w : 0.0f;
  }
  float s = (acc.x + acc.y) + (acc.z + acc.w);
  // All lanes reconverged here: EXEC is all ones for the WMMA.
  float ws = wave_sum_wmma(s);
  const int wave = tid >> 5;
  if ((tid & 31) == 0) smem[wave] = ws;
  __syncthreads();
  if (tid == 0) {
    float t = 0.0f;
    #pragma unroll
    for (int w = 0; w < 8; ++w) t += smem[w];
    partials[blockIdx.x] = t;
  }
}

// Pass 2: single block folds the block partials and applies the reference's
// scaling in the same order: loss = total / (I*(J-1)*K); out = THETA * loss.
__global__ void SparsityLoss_final_kernel(const float* __restrict__ partials,
                                          int n,
                                          float* __restrict__ out,
                                          float divisor, float theta) {
  __shared__ float smem[8];
  const int tid = threadIdx.x;
  float s = 0.0f;
  for (int i = tid; i < n; i += blockDim.x) s += partials[i];  // uniform trip count
  float ws = wave_sum_wmma(s);
  const int wave = tid >> 5;
  if ((tid & 31) == 0) smem[wave] = ws;
  __syncthreads();
  if (tid == 0) {
    float total = 0.0f;
    #pragma unroll
    for (int w = 0; w < 8; ++w) total += smem[w];
    float loss = total / divisor;
    out[0] = theta * loss;
  }
}

extern "C" void kernel_launch(void* const* d_in, const int* in_sizes, int n_in,
                              void* d_out, int out_size, void* d_ws, size_t ws_size,
                              hipStream_t stream) {
  (void)n_in; (void)out_size; (void)ws_size;
  const float* masks = (const float*)d_in[0];
  const unsigned char* valid = (const unsigned char*)d_in[1];
  float* out = (float*)d_out;
  float* partials = (float*)d_ws;

  const int n = in_sizes[0];        // 128*49*4096 = 25,690,112 (divisible by 4)
  const int n4 = n >> 2;            // float4 / uchar4 groups

  const int threads = 256;          // 8 waves per block on wave32
  const int nblocks = 2048;         // saturate HBM; 8 KB of partials in d_ws

  SparsityLoss_partial_kernel<<<nblocks, threads, 0, stream>>>(
      (const float4*)masks, (const uchar4*)valid, partials, n4);

  const float divisor = 128.0f * 48.0f * 4096.0f;  // I*(J-1)*K = 25,165,824 (exact in f32)
  const float theta = 1.0e-4f;
  SparsityLoss_final_kernel<<<1, threads, 0, stream>>>(
      partials, nblocks, out, divisor, theta);
}